// Network_49589692400277
// MI455X (gfx1250) — compile-verified
//
#include <hip/hip_runtime.h>
#include <hip/hip_bf16.h>
#include <math.h>

// ---------------------------------------------------------------------------
// Problem dims (fixed by the reference)
// ---------------------------------------------------------------------------
#define BB 16
#define LL 1024
#define DD 2048
#define HH 512
#define G4 2048          // 4*H
#define ML (BB * LL)     // 16384 rows of the big GEMM

#define ASTRIDE 132      // K1 LDS row stride (dwords): 132%64=4 -> conflict-free
#define WSTRIDE 516      // K2 LDS row stride (dwords): 16B-aligned, 516%64=4

typedef float    v2f __attribute__((ext_vector_type(2)));
typedef float    v8f __attribute__((ext_vector_type(8)));
typedef unsigned v4u __attribute__((ext_vector_type(4)));
typedef unsigned v8u __attribute__((ext_vector_type(8)));

__device__ __forceinline__ float sigmoidf_(float x) {
    return 1.0f / (1.0f + __expf(-x));
}

// Low 32 bits of a generic shared-memory address == wave-relative LDS byte
// offset on AMDGPU (aperture lives in the high 32 bits).
__device__ __forceinline__ unsigned lds_off(const void* p) {
    return (unsigned)(size_t)p;
}

// ---------------------------------------------------------------------------
// K0: zero h double-buffer + barrier counter (harness does not re-zero ws)
// ---------------------------------------------------------------------------
__global__ void k_init(float* __restrict__ h_all, unsigned* __restrict__ bar) {
    int i = blockIdx.x * blockDim.x + threadIdx.x;
    if (i < 2 * BB * HH) h_all[i] = 0.0f;
    if (i == 0) *bar = 0u;
}

// ---------------------------------------------------------------------------
// TDM descriptor issue: load a 16-row x 128-col fp32 tile of x (row stride
// 2048 elements) into LDS at byte offset `loff`, padding +4 dwords after
// every 128 dwords (-> LDS row stride 132 dwords). Groups 2/3 omitted (2D).
// ---------------------------------------------------------------------------
__device__ __forceinline__ void tdm_load_tile(const float* gsrc, unsigned loff) {
    unsigned long long ga = (unsigned long long)(size_t)gsrc;
    v4u g0;
    g0.x = 1u;                                   // count=1, user mode
    g0.y = loff;                                 // lds_addr (bytes)
    g0.z = (unsigned)(ga & 0xffffffffu);         // global_addr[31:0]
    g0.w = (unsigned)((ga >> 32) & 0x01ffffffu)  // global_addr[56:32]
         | (2u << 30);                           // type=2 ("image")
    v8u g1;
    g1.s0 = (2u << 16)                           // data_size = 4B
          | (1u << 20)                           // pad_enable
          | (6u << 22)                           // pad_interval: 2^(6+1)=128 dw
          | (3u << 25);                          // pad_amount: 3+1 = 4 dwords
    g1.s1 = (2048u << 16);                       // tensor_dim0 = 2048 (lo16)
    g1.s2 = ((unsigned)ML << 16);                // tensor_dim1 = 16384 (lo16)
    g1.s3 = (128u << 16);                        // tile_dim0 = 128
    g1.s4 = 16u;                                 // tile_dim1 = 16, tile_dim2 = 0
    g1.s5 = 2048u;                               // tensor_dim0_stride[31:0]
    g1.s6 = 0u;
    g1.s7 = 0u;
    asm volatile("tensor_load_to_lds %0, %1" :: "s"(g0), "s"(g1) : "memory");
}

// ---------------------------------------------------------------------------
// K1: xg[M=16384, N=2048] = x[M, K=2048] @ W_ih^T + (b_ih + b_hh)
// fp32 WMMA 16x16x4. Block = 8 waves, one M-tile per block, wave w owns
// N-tile (blockIdx.y*8 + w). A K-chunks staged into LDS by the Tensor Data
// Mover (double buffered, s_wait_tensorcnt); B streamed from L2 (W_ih is
// 16 MB, fully L2-resident) with prefetch of the next chunk.
//
// Fragment layouts (ISA 7.12.2, 32-bit, wave32):
//   A 16x4 : lane<16 -> M=lane,    VGPR0:K=k0+0 ; lane>=16 -> M=lane-16, K=k0+2
//   B  4x16: same with M->N (contiguous K pair per lane -> float2 loads)
//   C/D    : VGPR r -> M = r + 8*(lane/16), N = lane%16
// ---------------------------------------------------------------------------
__global__ void __launch_bounds__(256)
k_gemm_wih(const float* __restrict__ x, const float* __restrict__ Wih,
           const float* __restrict__ b_ih, const float* __restrict__ b_hh,
           float* __restrict__ xg) {
    __shared__ __attribute__((aligned(16))) float As[2][16 * ASTRIDE];

    const int tid  = threadIdx.x;
    const int wave = tid >> 5;
    const int lane = tid & 31;
    const int lm   = lane & 15;          // row/col within tile
    const int hi   = lane >> 4;          // K half-select
    const int m0   = blockIdx.x * 16;
    const int n0   = (blockIdx.y * 8 + wave) * 16;

    v8f acc = {0.f, 0.f, 0.f, 0.f, 0.f, 0.f, 0.f, 0.f};

    if (tid == 0)
        tdm_load_tile(x + (size_t)m0 * DD, lds_off(&As[0][0]));

    int buf = 0;
    for (int kc = 0; kc < DD; kc += 128, buf ^= 1) {
        if (kc + 128 < DD) {
            if (tid == 0)
                tdm_load_tile(x + (size_t)m0 * DD + (kc + 128),
                              lds_off(&As[buf ^ 1][0]));
            __builtin_amdgcn_s_wait_tensorcnt((short)1);  // prev tile done
        } else {
            __builtin_amdgcn_s_wait_tensorcnt((short)0);
        }
        __syncthreads();

        const float* brow = Wih + (size_t)(n0 + lm) * DD + kc;
        if (kc + 128 < DD)
            __builtin_prefetch(brow + 128, 0, 3);  // next B chunk -> L2/L0
        const float* arow = &As[buf][0] + lm * ASTRIDE;
#pragma unroll 8
        for (int k = 0; k < 128; k += 4) {
            const int kk = k + 2 * hi;
            v2f a = *(const v2f*)(arow + kk);
            v2f b = *(const v2f*)(brow + kk);
            acc = __builtin_amdgcn_wmma_f32_16x16x4_f32(
                false, a, false, b, (short)0, acc, false, false);
        }
        __syncthreads();   // all waves done with As[buf] before TDM refills it
    }

    const int   col  = n0 + lm;
    const float bias = b_ih[col] + b_hh[col];
#pragma unroll
    for (int r = 0; r < 8; ++r) {
        const int row = m0 + r + 8 * hi;
        xg[(size_t)row * G4 + col] = acc[r] + bias;
    }
}

// ---------------------------------------------------------------------------
// K2: LSTM scan. 16 persistent WGs; WG j owns h-slice [32j, 32j+32) and the
// 128 matching W_hh rows (i/f/g/o), resident in LDS (128 x 516 floats),
// staged via async global->LDS B128 copies (ASYNCcnt). Per step:
// gates[16b x 128] = h[16,512] @ Wslice^T via 8 waves x 128 chained f32
// WMMAs, + xg, elementwise cell update, h written to the other half of a
// double-buffered global h, then a device-scope atomic barrier.
// ---------------------------------------------------------------------------
__global__ void __launch_bounds__(256)
k_lstm_scan(const float* __restrict__ xg, const float* __restrict__ Whh,
            float* __restrict__ h_all,   // [2][16][512] double buffer
            float* __restrict__ lstm_out,
            unsigned* __restrict__ bar) {
    extern __shared__ __attribute__((aligned(16))) float sm[];
    float* Wl    = sm;                        // 128 * WSTRIDE
    float* gates = sm + 128 * WSTRIDE;        // 16 * 128
    float* cbuf  = gates + 16 * 128;          // 16 * 32

    const int j    = blockIdx.x;              // h-slice 0..15
    const int tid  = threadIdx.x;
    const int wave = tid >> 5;
    const int lane = tid & 31;
    const int lm   = lane & 15;
    const int hi   = lane >> 4;
    const int nl   = wave * 16 + lm;          // gate column-local 0..127

    // Async-stage W_hh slice: rows {q*512 + 32j + rr}, q=0..3, rr=0..31.
    // 128 rows x 128 b128-units; per-lane 16B transfers, no VGPR round trip.
    {
        const unsigned wl0 = lds_off(Wl);
        for (int u = tid; u < 128 * 128; u += 256) {
            const int r  = u >> 7;            // LDS row 0..127
            const int c4 = (u & 127) << 2;    // column (floats), 16B units
            const int q  = r >> 5, rr = r & 31;
            const float* gp =
                Whh + (size_t)(q * HH + 32 * j + rr) * HH + c4;
            const unsigned loff = wl0 + (unsigned)((r * WSTRIDE + c4) * 4);
            asm volatile("global_load_async_to_lds_b128 %0, %1, off"
                         :: "v"(loff), "v"(gp) : "memory");
        }
    }
    for (int i = tid; i < BB * 32; i += 256) cbuf[i] = 0.0f;
    asm volatile("s_wait_asynccnt 0x0" ::: "memory");
    __syncthreads();

    // gate-global column for this lane's D-fragment column
    const int gc = (nl >> 5) * HH + 32 * j + (nl & 31);

    for (int t = 0; t < LL; ++t) {
        const float* hrd = h_all + (t & 1) * (BB * HH);
        float*       hwr = h_all + ((t + 1) & 1) * (BB * HH);

        v8f acc = {0.f, 0.f, 0.f, 0.f, 0.f, 0.f, 0.f, 0.f};
        const float* arow = hrd + lm * HH;
        const float* brow = Wl + nl * WSTRIDE;
#pragma unroll 4
        for (int k = 0; k < HH; k += 4) {
            const int kk = k + 2 * hi;
            v2f a = *(const v2f*)(arow + kk);
            v2f b = *(const v2f*)(brow + kk);
            acc = __builtin_amdgcn_wmma_f32_16x16x4_f32(
                false, a, false, b, (short)0, acc, false, false);
        }

        // pre-activations -> LDS (+ precomputed input projection)
#pragma unroll
        for (int r = 0; r < 8; ++r) {
            const int m = r + 8 * hi;                       // batch index
            gates[m * 128 + nl] =
                acc[r] + xg[(size_t)(m * LL + t) * G4 + gc];
        }
        __syncthreads();

        // elementwise cell update: 16 batches x 32 channels = 512 items
        for (int e = tid; e < BB * 32; e += 256) {
            const int b_ = e >> 5, hl = e & 31;
            const float gi = gates[b_ * 128 + hl];
            const float gf = gates[b_ * 128 + 32 + hl];
            const float gg = gates[b_ * 128 + 64 + hl];
            const float go = gates[b_ * 128 + 96 + hl];
            float c = cbuf[e];
            c = sigmoidf_(gf) * c + sigmoidf_(gi) * tanhf(gg);
            const float h = sigmoidf_(go) * tanhf(c);
            cbuf[e] = c;
            const int hg = 32 * j + hl;
            hwr[b_ * HH + hg] = h;
            lstm_out[(size_t)(b_ * LL + t) * HH + hg] = h;
        }

        // device-scope barrier (monotonic counter; zeroed by k_init)
        __threadfence();
        __syncthreads();
        if (tid == 0)
            __hip_atomic_fetch_add(bar, 1u, __ATOMIC_ACQ_REL,
                                   __HIP_MEMORY_SCOPE_AGENT);
        while (__hip_atomic_load(bar, __ATOMIC_ACQUIRE,
                                 __HIP_MEMORY_SCOPE_AGENT) < 16u * (t + 1))
            __builtin_amdgcn_s_sleep(1);
        __syncthreads();
    }
}

// ---------------------------------------------------------------------------
// K3a: per-position argmax gate. One wave per (b,t); wave32 shfl reduction.
// ---------------------------------------------------------------------------
__global__ void __launch_bounds__(256)
k_classify(const float* __restrict__ lstm, const float* __restrict__ Wlin,
           const float* __restrict__ blin, float* __restrict__ g) {
    const int tid  = threadIdx.x;
    const int wave = tid >> 5;
    const int lane = tid & 31;
    const int p    = blockIdx.x * 8 + wave;      // 0..16383
    const float* v = lstm + (size_t)p * HH;
    float l0 = 0.f, l1 = 0.f;
#pragma unroll
    for (int i = lane; i < HH; i += 32) {
        const float xv = v[i];
        l0 += xv * Wlin[i];
        l1 += xv * Wlin[HH + i];
    }
    for (int o = 16; o > 0; o >>= 1) {
        l0 += __shfl_xor(l0, o, 32);
        l1 += __shfl_xor(l1, o, 32);
    }
    if (lane == 0)
        g[p] = (l1 + blin[1] > l0 + blin[0]) ? 1.0f : 0.0f;
}

// ---------------------------------------------------------------------------
// K3b: out = 2*lstm + exclusive-prefix(g*lstm*fwd) + exclusive-suffix(g*lstm*bwd)
// One WG per batch, one channel per thread; two sequential sweeps over t.
// ---------------------------------------------------------------------------
__global__ void __launch_bounds__(512)
k_fuse(const float* __restrict__ lstm, const float* __restrict__ g,
       const float* __restrict__ fatt, const float* __restrict__ batt,
       float* __restrict__ out) {
    const int b  = blockIdx.x;
    const int hl = threadIdx.x;                  // 0..511
    const float fa = fatt[hl], ba = batt[hl];
    const float* Lb = lstm + (size_t)b * LL * HH;
    const float* gb = g + b * LL;
    float*       ob = out + (size_t)b * LL * HH;

    float acc = 0.f;
    for (int t = 0; t < LL; ++t) {
        const float lv = Lb[(size_t)t * HH + hl];
        ob[(size_t)t * HH + hl] = 2.0f * lv + acc;
        acc += gb[t] * lv * fa;
    }
    acc = 0.f;
    for (int t = LL - 1; t >= 0; --t) {
        const float lv = Lb[(size_t)t * HH + hl];
        ob[(size_t)t * HH + hl] += acc;
        acc += gb[t] * lv * ba;
    }
}

// ---------------------------------------------------------------------------
// launch
// ---------------------------------------------------------------------------
extern "C" void kernel_launch(void* const* d_in, const int* in_sizes, int n_in,
                              void* d_out, int out_size, void* d_ws, size_t ws_size,
                              hipStream_t stream) {
    (void)in_sizes; (void)n_in; (void)out_size; (void)ws_size;

    const float* x     = (const float*)d_in[0];
    const float* W_ih  = (const float*)d_in[1];
    const float* W_hh  = (const float*)d_in[2];
    const float* b_ih  = (const float*)d_in[3];
    const float* b_hh  = (const float*)d_in[4];
    const float* W_lin = (const float*)d_in[5];
    const float* b_lin = (const float*)d_in[6];
    const float* f_att = (const float*)d_in[7];
    const float* b_att = (const float*)d_in[8];
    float*       out   = (float*)d_out;

    char* ws = (char*)d_ws;
    float*    xg    = (float*)(ws);                           // 128 MB
    float*    lstm  = (float*)(ws + (size_t)134217728);       //  32 MB
    float*    h_all = (float*)(ws + (size_t)167772160);       //  64 KB (2 bufs)
    float*    gbuf  = (float*)(ws + (size_t)167837696);       //  64 KB
    unsigned* bar   = (unsigned*)(ws + (size_t)167903232);    //   4 B

    // K2 dynamic LDS: (128*516 + 16*128 + 512) floats = 274432 B (<320KB/WGP)
    const int smem2 = (128 * WSTRIDE + 16 * 128 + 512) * (int)sizeof(float);
    hipFuncSetAttribute((const void*)k_lstm_scan,
                        hipFuncAttributeMaxDynamicSharedMemorySize, smem2);

    k_init<<<64, 256, 0, stream>>>(h_all, bar);

    dim3 g1(ML / 16, G4 / (16 * 8));                          // (1024, 16)
    k_gemm_wih<<<g1, 256, 0, stream>>>(x, W_ih, b_ih, b_hh, xg);

    k_lstm_scan<<<16, 256, smem2, stream>>>(xg, W_hh, h_all, lstm, bar);

    k_classify<<<ML / 8, 256, 0, stream>>>(lstm, W_lin, b_lin, gbuf);

    k_fuse<<<BB, HH, 0, stream>>>(lstm, gbuf, f_att, b_att, out);
}